// VehicleSelectionDecoder_10007273799966
// MI455X (gfx1250) — compile-verified
//
#include <hip/hip_runtime.h>
#include <math.h>
#include <stdint.h>

// ---------------------------------------------------------------------------
// VehicleSelectionDecoder on gfx1250 (MI455X), fp32 end-to-end.
// One block = 256 threads (8 wave32) = 4 batch elements.
// - Dominant GEMM (h = [feat|loc] @ Wvp.T) on V_WMMA_F32_16X16X4_F32.
// - Input streaming via GLOBAL_LOAD_ASYNC_TO_LDS (ASYNCcnt path, no VGPRs).
// - Attention K/V/K2 projections algebraically folded into the query side,
//   so only h (64x128 per batch element) is ever materialized (in LDS).
// ---------------------------------------------------------------------------

constexpr int kE  = 128;   // embed dim
constexpr int kV  = 64;    // vehicles
constexpr int kH  = 8;     // heads
constexpr int kF  = 5;     // vehicle feature count
constexpr int kIn = kF + kE;  // 133 = Wvp input dim
constexpr int kBT = 4;     // batch elements per block
constexpr int kThreads = 256;
constexpr int kHP = kE + 4;   // padded LDS row stride (bank-conflict avoidance)

typedef __attribute__((ext_vector_type(2))) float v2f;
typedef __attribute__((ext_vector_type(8))) float v8f;

// LDS layout (in floats), dynamic shared memory
constexpr int OFF_H    = 0;                           // [kBT*kV][kHP]  loc staging, then h
constexpr int OFF_BT   = OFF_H    + kBT * kV * kHP;   // [kE][kHP]      Wvp.T (e-part), frag layout
constexpr int OFF_FEAT = OFF_BT   + kE * kHP;         // [kBT*kV*kF]
constexpr int OFF_G    = OFF_FEAT + kBT * kV * kF;    // [kBT*kE]
constexpr int OFF_QP   = OFF_G    + kBT * kE;         // [kBT*kE]
constexpr int OFF_QK   = OFF_QP   + kBT * kE;         // [kBT*kH*kE]
constexpr int OFF_SB   = OFF_QK   + kBT * kH * kE;    // [kBT*kH]
constexpr int OFF_ATTN = OFF_SB   + kBT * kH;         // [kBT*kH*kV]
constexpr int OFF_HBAR = OFF_ATTN + kBT * kH * kV;    // [kBT*kH*kE]
constexpr int OFF_CTX  = OFF_HBAR + kBT * kH * kE;    // [kBT*kE]
constexpr int OFF_ATT  = OFF_CTX  + kBT * kE;         // [kBT*kE]
constexpr int OFF_Q2   = OFF_ATT  + kBT * kE;         // [kBT*kE]
constexpr int OFF_QF   = OFF_Q2   + kBT * kE;         // [kBT*kE]
constexpr int SMEM_F   = OFF_QF   + kBT * kE;         // total floats (~65312 -> ~255 KB)

// Async global->LDS copy helpers (gfx1250; tracked by ASYNCcnt).
// the LDS byte address is taken from the vdst VGPR; low 32 bits of a generic LDS pointer are the
// wave-relative LDS byte offset (aperture carries only the high 32 bits).
__device__ __forceinline__ void async_copy_b128(void* lds_dst, const void* gsrc) {
  const uint32_t ldsa = (uint32_t)(uintptr_t)lds_dst;
  const uint64_t ga   = (uint64_t)(uintptr_t)gsrc;
  asm volatile("global_load_async_to_lds_b128 %0, %1, off"
               :: "v"(ldsa), "v"(ga) : "memory");
}
__device__ __forceinline__ void async_copy_b32(void* lds_dst, const void* gsrc) {
  const uint32_t ldsa = (uint32_t)(uintptr_t)lds_dst;
  const uint64_t ga   = (uint64_t)(uintptr_t)gsrc;
  asm volatile("global_load_async_to_lds_b32 %0, %1, off"
               :: "v"(ldsa), "v"(ga) : "memory");
}
__device__ __forceinline__ void wait_async_zero() {
  asm volatile("s_wait_asynccnt 0x0" ::: "memory");
}

__launch_bounds__(kThreads, 1)
__global__ void vsd_kernel(
    const float* __restrict__ g,     // [B,E] graph embedding
    const float* __restrict__ feat,  // [B,V,F]
    const float* __restrict__ loc,   // [B,V,E]
    const int*   __restrict__ mask,  // [B,V] nonzero => masked out
    const float* __restrict__ Wvp,   // [E, F+E]
    const float* __restrict__ bvp,   // [E]
    const float* __restrict__ Wio,   // [3E, E] packed q,k,v
    const float* __restrict__ bio,   // [3E]
    const float* __restrict__ Wout,  // [E,E]
    const float* __restrict__ bout,  // [E]
    const float* __restrict__ Wq2,   // [E,E]
    const float* __restrict__ Wk2,   // [E,E]
    float* __restrict__ out)         // [B,V]
{
  extern __shared__ float smem[];
  float* sh_h    = smem + OFF_H;
  float* sh_Bt   = smem + OFF_BT;
  float* sh_feat = smem + OFF_FEAT;
  float* sh_g    = smem + OFF_G;
  float* sh_qp   = smem + OFF_QP;
  float* sh_qk   = smem + OFF_QK;
  float* sh_sb   = smem + OFF_SB;
  float* sh_attn = smem + OFF_ATTN;
  float* sh_hbar = smem + OFF_HBAR;
  float* sh_ctx  = smem + OFF_CTX;
  float* sh_att  = smem + OFF_ATT;
  float* sh_q2   = smem + OFF_Q2;
  float* sh_qf   = smem + OFF_QF;

  const int t    = threadIdx.x;
  const int lane = t & 31;
  const int wave = t >> 5;
  const int b0   = blockIdx.x * kBT;   // global batch base for this block

  // ---------------- Stage inputs into LDS (async global->LDS) ----------------
  // loc -> sh_h rows (padded stride kHP), b128 async copies, coalesced
  {
    const float* src = loc + (size_t)b0 * kV * kE;
    const int nvec = kBT * kV * kE / 4;  // 8192 16-byte chunks
    for (int idx = t; idx < nvec; idx += kThreads) {
      const int row  = idx >> 5;        // 32 float4 per 128-float row
      const int col4 = idx & 31;
      async_copy_b128(sh_h + row * kHP + col4 * 4, src + (size_t)idx * 4);
    }
  }
  // Wvp.T (embedding part): sh_Bt[n*kHP + k] = Wvp[n*133 + 5 + k]
  // (row pitch 133 floats -> no 16B alignment; stream b32, coalesced over k)
  for (int idx = t; idx < kE * kE; idx += kThreads) {
    const int n = idx >> 7;
    const int k = idx & 127;
    async_copy_b32(sh_Bt + n * kHP + k, Wvp + n * kIn + kF + k);
  }
  for (int idx = t; idx < kBT * kV * kF; idx += kThreads)
    sh_feat[idx] = feat[(size_t)b0 * kV * kF + idx];
  for (int idx = t; idx < kBT * kE; idx += kThreads)
    sh_g[idx] = g[(size_t)b0 * kE + idx];
  wait_async_zero();
  __syncthreads();

  // ---------------- GEMM1: h = loc @ Wvp_e.T  (WMMA f32 16x16x4) -------------
  // M = kBT*kV = 256 rows, N = 128, K = 128. Wave w owns M-tiles {2w, 2w+1}.
  // Fragment layout (ISA 7.12.2, 32-bit):
  //   A 16x4 : lanes 0-15 -> {K0,K1}, lanes 16-31 -> {K2,K3}, row = lane%16
  //   B 4x16 : lanes 0-15 -> {K0,K1}, lanes 16-31 -> {K2,K3}, col = lane%16
  //   C/D    : vgpr j -> row j (lanes 0-15) / row 8+j (lanes 16-31), col = lane%16
  {
    const int lh   = lane & 15;
    const int koff = (lane >> 4) << 1;       // 0 or 2
    const int rsel = (lane >> 4) << 3;       // 0 or 8 (C/D row half)
    for (int mi = 0; mi < 2; ++mi) {
      const int mt = wave * 2 + mi;
      const float* Arow = sh_h + (mt * 16 + lh) * kHP + koff;
      v8f acc[8];
      #pragma unroll
      for (int nt = 0; nt < 8; ++nt)
        acc[nt] = (v8f){0.f, 0.f, 0.f, 0.f, 0.f, 0.f, 0.f, 0.f};
      #pragma unroll 2
      for (int k0 = 0; k0 < kE; k0 += 4) {
        const v2f a = *(const v2f*)(Arow + k0);
        #pragma unroll
        for (int nt = 0; nt < 8; ++nt) {
          const v2f b = *(const v2f*)(sh_Bt + (nt * 16 + lh) * kHP + k0 + koff);
          acc[nt] = __builtin_amdgcn_wmma_f32_16x16x4_f32(
              false, a, false, b, (short)0, acc[nt], false, false);
        }
      }
      // Epilogue: bias + rank-5 feature update, then h overwrites loc in LDS.
      // (Same-wave DS ordering: all A reads of these rows precede the stores.)
      #pragma unroll
      for (int nt = 0; nt < 8; ++nt) {
        const int n = nt * 16 + lh;
        const float bias = bvp[n];
        const float w0 = Wvp[n * kIn + 0], w1 = Wvp[n * kIn + 1],
                    w2 = Wvp[n * kIn + 2], w3 = Wvp[n * kIn + 3],
                    w4 = Wvp[n * kIn + 4];
        #pragma unroll
        for (int j = 0; j < 8; ++j) {
          const int m = mt * 16 + rsel + j;        // flat row in [0, kBT*kV)
          const float* fp = sh_feat + m * kF;
          const float hv = acc[nt][j] + bias + fp[0] * w0 + fp[1] * w1 +
                           fp[2] * w2 + fp[3] * w3 + fp[4] * w4;
          sh_h[m * kHP + n] = hv;
        }
      }
    }
  }
  __syncthreads();

  // ---------------- qp = Wq @ g + bq ----------------
  #pragma unroll
  for (int i = 0; i < 2; ++i) {
    const int idx = i * kThreads + t;          // kBT*kE = 512 outputs
    const int b = idx >> 7, o = idx & 127;
    const float* wr = Wio + (size_t)o * kE;
    const float* gg = sh_g + b * kE;
    float s = bio[o];
    #pragma unroll 4
    for (int e = 0; e < kE; ++e) s += wr[e] * gg[e];
    sh_qp[idx] = s;
  }
  __syncthreads();

  // ---------------- fold K-projection into query: qk_h = Wk_h^T qp_h --------
  #pragma unroll
  for (int i = 0; i < 16; ++i) {
    const int idx = i * kThreads + t;          // kBT*kH*kE = 4096 outputs
    const int b = idx >> 10, hh = (idx >> 7) & 7, e = idx & 127;
    const float* wb  = Wio + (size_t)(kE + hh * 16) * kE + e;  // Wk rows
    const float* qpp = sh_qp + b * kE + hh * 16;
    float s = 0.f;
    #pragma unroll
    for (int d = 0; d < 16; ++d) s += wb[d * kE] * qpp[d];
    sh_qk[idx] = s;
  }
  if (t < kBT * kH) {                          // sb_h = qp_h . bk_h
    const int b = t >> 3, hh = t & 7;
    float s = 0.f;
    #pragma unroll
    for (int d = 0; d < 16; ++d)
      s += sh_qp[b * kE + hh * 16 + d] * bio[kE + hh * 16 + d];
    sh_sb[t] = s;
  }
  __syncthreads();

  // ---------------- scores + mask ----------------
  #pragma unroll
  for (int i = 0; i < 8; ++i) {
    const int idx = i * kThreads + t;          // kBT*kH*kV = 2048 outputs
    const int b = idx >> 9, hh = (idx >> 6) & 7, vv = idx & 63;
    const float* hr  = sh_h + (b * kV + vv) * kHP;
    const float* qkr = sh_qk + (b * kH + hh) * kE;
    float s = 0.f;
    #pragma unroll 4
    for (int e = 0; e < kE; ++e) s += hr[e] * qkr[e];
    s = (s + sh_sb[b * kH + hh]) * 0.25f;      // 1/sqrt(D), D=16
    if (mask[(size_t)(b0 + b) * kV + vv]) s = -INFINITY;
    sh_attn[idx] = s;
  }
  __syncthreads();

  // ---------------- softmax over v, per (b, head) ----------------
  if (t < kBT * kH) {
    float* row = sh_attn + t * kV;
    float m = -INFINITY;
    for (int v = 0; v < kV; ++v) m = fmaxf(m, row[v]);
    float sum = 0.f;
    for (int v = 0; v < kV; ++v) { const float ex = __expf(row[v] - m); row[v] = ex; sum += ex; }
    const float inv = 1.f / sum;
    for (int v = 0; v < kV; ++v) row[v] *= inv;
  }
  __syncthreads();

  // ---------------- hbar_h = attn_h^T @ h ----------------
  #pragma unroll
  for (int i = 0; i < 16; ++i) {
    const int idx = i * kThreads + t;          // kBT*kH*kE
    const int b = idx >> 10, hh = (idx >> 7) & 7, e = idx & 127;
    const float* ar = sh_attn + (b * kH + hh) * kV;
    const float* hb = sh_h + b * kV * kHP + e;
    float s = 0.f;
    #pragma unroll 4
    for (int v = 0; v < kV; ++v) s += ar[v] * hb[v * kHP];
    sh_hbar[idx] = s;
  }
  __syncthreads();

  // ---------------- ctx = Wv @ hbar + bv (per head) ----------------
  #pragma unroll
  for (int i = 0; i < 2; ++i) {
    const int idx = i * kThreads + t;
    const int b = idx >> 7, o = idx & 127, hh = o >> 4;
    const float* wr = Wio + (size_t)(2 * kE + o) * kE;   // Wv rows
    const float* hb = sh_hbar + (b * kH + hh) * kE;
    float s = bio[2 * kE + o];
    #pragma unroll 4
    for (int e = 0; e < kE; ++e) s += wr[e] * hb[e];
    sh_ctx[idx] = s;
  }
  __syncthreads();

  // ---------------- attended = out_w @ ctx + out_b ----------------
  #pragma unroll
  for (int i = 0; i < 2; ++i) {
    const int idx = i * kThreads + t;
    const int b = idx >> 7, o = idx & 127;
    const float* wr = Wout + (size_t)o * kE;
    const float* cc = sh_ctx + b * kE;
    float s = bout[o];
    #pragma unroll 4
    for (int e = 0; e < kE; ++e) s += wr[e] * cc[e];
    sh_att[idx] = s;
  }
  __syncthreads();

  // ---------------- q2 = Wq2 @ attended ----------------
  #pragma unroll
  for (int i = 0; i < 2; ++i) {
    const int idx = i * kThreads + t;
    const int b = idx >> 7, o = idx & 127;
    const float* wr = Wq2 + (size_t)o * kE;
    const float* aa = sh_att + b * kE;
    float s = 0.f;
    #pragma unroll 4
    for (int e = 0; e < kE; ++e) s += wr[e] * aa[e];
    sh_q2[idx] = s;
  }
  __syncthreads();

  // ---------------- fold K2: qf = Wk2^T @ q2 (coalesced over e) -------------
  #pragma unroll
  for (int i = 0; i < 2; ++i) {
    const int idx = i * kThreads + t;
    const int b = idx >> 7, e = idx & 127;
    const float* q2p = sh_q2 + b * kE;
    float s = 0.f;
    #pragma unroll 4
    for (int o = 0; o < kE; ++o) s += Wk2[(size_t)o * kE + e] * q2p[o];
    sh_qf[idx] = s;
  }
  __syncthreads();

  // ---------------- logits = tanh-clip( h . qf / sqrt(E) ), mask ------------
  {
    const int b = t >> 6, vv = t & 63;         // kBT*kV = 256 = one per thread
    const float* hr  = sh_h + (b * kV + vv) * kHP;
    const float* qfr = sh_qf + b * kE;
    float s = 0.f;
    #pragma unroll 4
    for (int e = 0; e < kE; ++e) s += hr[e] * qfr[e];
    s *= 0.08838834764831845f;                 // 1/sqrt(128)
    const float val = 10.f * tanhf(s);
    const size_t oi = (size_t)(b0 + b) * kV + vv;
    out[oi] = mask[oi] ? -INFINITY : val;
  }
}

extern "C" void kernel_launch(void* const* d_in, const int* in_sizes, int n_in,
                              void* d_out, int out_size, void* d_ws, size_t ws_size,
                              hipStream_t stream) {
  const float* g    = (const float*)d_in[0];
  const float* feat = (const float*)d_in[1];
  const float* loc  = (const float*)d_in[2];
  const int*   mask = (const int*)  d_in[3];
  const float* Wvp  = (const float*)d_in[4];
  const float* bvp  = (const float*)d_in[5];
  const float* Wio  = (const float*)d_in[6];
  const float* bio  = (const float*)d_in[7];
  const float* Wout = (const float*)d_in[8];
  const float* bout = (const float*)d_in[9];
  const float* Wq2  = (const float*)d_in[10];
  const float* Wk2  = (const float*)d_in[11];
  float* out = (float*)d_out;

  const int b = in_sizes[0] / kE;              // batch (8192)
  const int blocks = b / kBT;
  const size_t shmem = (size_t)SMEM_F * sizeof(float);  // ~255 KB (<= 320 KB/WGP)

  vsd_kernel<<<dim3(blocks), dim3(kThreads), shmem, stream>>>(
      g, feat, loc, mask, Wvp, bvp, Wio, bio, Wout, bout, Wq2, Wk2, out);
}